// linear_nce_57071525429754
// MI455X (gfx1250) — compile-verified
//
#include <hip/hip_runtime.h>

// NCE forward fused kernel for MI455X (gfx1250, wave32).
//
// Outputs (flat in d_out, reference return order):
//   pmt [N]   = exp(dot(input[n], W[target[n]]) + bias[target[n]])
//   pnt [N]   = unigram[target[n]]
//   pmn [N*K] = exp(input @ W[noise].T + bias[noise])   (row-major [N,K])
//   pnn [N*K] = broadcast unigram[noise]
//
// Memory-bound: ~272 MB total traffic -> ~11.7 us floor at 23.3 TB/s.
// GEMM rides V_WMMA_F32_16X16X4_F32; staging uses CDNA5 async-to-LDS
// (global_load_async_to_lds_b128 via inline asm, ASYNCcnt) with double
// buffering so chunk c+1 streams from HBM while chunk c's WMMAs run.

typedef float v2f __attribute__((ext_vector_type(2)));
typedef float v8f __attribute__((ext_vector_type(8)));

#define N_TOK 16384
#define HID   2048
#define KNZ   25
#define KPAD  32

#define TM    64            // tokens per workgroup
#define BLK   128           // threads per workgroup (4 waves)
#define HC    64            // H-chunk staged in LDS
#define NCH   (HID / HC)    // 32 chunks
#define LPAD  68            // LDS row stride (floats): 272B rows -> 16B aligned,
                            // 68 % 64 = 4 banks/row -> conflict-free WMMA frag reads

#define AS3 __attribute__((address_space(3)))

// Wave-relative 32-bit LDS byte offset of a generic shared-memory pointer.
static __device__ __forceinline__ unsigned lds_off(const void* p) {
    return (unsigned)(unsigned long long)(AS3 const void*)p;
}

// One 16B per-lane async copy: from global address g to LDS address l, tracked by ASYNCcnt.
// VGLOBAL async encoding: VDST = LDS address VGPR, VADDR = 64-bit global
// address, SADDR = NULL ("off").  (CDNA5 ISA)
static __device__ __forceinline__ void async_copy16(const void* g, void* l) {
    asm volatile("global_load_async_to_lds_b128 %0, %1, off"
                 :
                 : "v"(lds_off(l)), "v"(g)
                 : "memory");
}

__global__ __launch_bounds__(BLK) void nce_fused_kernel(
    const float* __restrict__ input,    // [N, H]
    const float* __restrict__ weight,   // [V, H]
    const float* __restrict__ bias,     // [V]
    const float* __restrict__ uni,      // [V]
    const int*   __restrict__ target,   // [N]
    const int*   __restrict__ noise,    // [K]
    float* __restrict__ pmt,
    float* __restrict__ pnt,
    float* __restrict__ pmn,
    float* __restrict__ pnn)
{
    __shared__ float inTile[2][TM][LPAD];
    __shared__ float nzTile[2][KPAD][LPAD];

    const int tid       = threadIdx.x;
    const int lane      = tid & 31;
    const int wv        = tid >> 5;                // wave id 0..3
    const int blockTok0 = blockIdx.x * TM;
    const int tokBaseW  = blockTok0 + wv * 16;     // this wave's 16-token tile

    // Per-lane WMMA column assignment (two 16-wide N-tiles covering KPAD=32)
    const int half = lane >> 4;                    // 0/1: K-offset half of the 4-step
    const int lcol = lane & 15;
    const int c0   = lcol;                         // 0..15  (always < KNZ)
    const int c1   = lcol + 16;                    // 16..31 (valid iff < KNZ)
    const int nIdx0 = noise[c0];
    const int nIdx1 = noise[c1 < KNZ ? c1 : (KNZ - 1)];
    const float bn0 = bias[nIdx0], un0 = uni[nIdx0];
    const float bn1 = bias[nIdx1], un1 = uni[nIdx1];
    const int kOff = half * 2;                     // A/B k offset within one 16x16x4 step

    // Staging coordinates (block-cooperative, 256B-contiguous b128 segments)
    const int r0 = tid >> 4;                       // 0..7 : staging row group
    const int cv = tid & 15;                       // float4 column within chunk

    // Hoist noise row indices so async issues don't stall on index loads
    int nzrow[KPAD / 8];
    #pragma unroll
    for (int i = 0; i < KPAD / 8; ++i) {
        const int row = r0 + i * 8;
        nzrow[i] = noise[row < KNZ ? row : (KNZ - 1)];
    }

    // Gathered target rows for this wave's 16 tokens
    int tgt[16];
    const float* wrow[16];
    #pragma unroll
    for (int t = 0; t < 16; ++t) {
        tgt[t]  = target[tokBaseW + t];
        wrow[t] = weight + (size_t)tgt[t] * HID;
    }

    v8f acc0 = {};
    v8f acc1 = {};
    float pacc[16];
    #pragma unroll
    for (int t = 0; t < 16; ++t) pacc[t] = 0.f;

    // ---- async staging of one H-chunk into LDS buffer `buf` (12 ops/wave) ----
    auto stage = [&](int chunk, int buf) {
        const int h0 = chunk * HC;
        #pragma unroll
        for (int i = 0; i < TM / 8; ++i) {         // 8 x b128 per thread
            const int row = r0 + i * 8;
            async_copy16(input + (size_t)(blockTok0 + row) * HID + h0 + cv * 4,
                         &inTile[buf][row][cv * 4]);
        }
        #pragma unroll
        for (int i = 0; i < KPAD / 8; ++i) {       // 4 x b128 per thread
            const int row = r0 + i * 8;
            async_copy16(weight + (size_t)nzrow[i] * HID + h0 + cv * 4,
                         &nzTile[buf][row][cv * 4]);
        }
    };

    // Prologue: fill both buffers
    stage(0, 0);
    stage(1, 1);

    for (int c = 0; c < NCH; ++c) {
        const int cur = c & 1;
        const int h0  = c * HC;

        // Chunk c's async copies landed (in-order completion; at most the
        // 12 copies of chunk c+1 may remain outstanding).
        if (c == NCH - 1) __builtin_amdgcn_s_wait_asynccnt(0);
        else              __builtin_amdgcn_s_wait_asynccnt(12);
        __syncthreads();                           // all waves see buf[cur] filled

        // ---- WMMA: acc += A(16 x HC) * B(HC x 32), in 16x16x4 f32 steps ----
        const int ar = wv * 16 + lcol;             // A row (M) for this lane
        #pragma unroll
        for (int kk = 0; kk < HC; kk += 4) {
            v2f a, b0, b1;
            a.x  = inTile[cur][ar][kk + kOff];
            a.y  = inTile[cur][ar][kk + kOff + 1];
            b0.x = nzTile[cur][c0][kk + kOff];
            b0.y = nzTile[cur][c0][kk + kOff + 1];
            b1.x = nzTile[cur][c1][kk + kOff];
            b1.y = nzTile[cur][c1][kk + kOff + 1];
            acc0 = __builtin_amdgcn_wmma_f32_16x16x4_f32(
                       false, a, false, b0, (short)0, acc0, false, false);
            acc1 = __builtin_amdgcn_wmma_f32_16x16x4_f32(
                       false, a, false, b1, (short)0, acc1, false, false);
        }

        // ---- pmt partials: dot(input row, gathered target row), this chunk ----
        // Per-lane float2 covers the 64-float chunk exactly (32 lanes * 2).
        #pragma unroll
        for (int t = 0; t < 16; ++t) {
            const float2 wval = *(const float2*)(wrow[t] + h0 + lane * 2);
            const float2 ival = *(const float2*)&inTile[cur][wv * 16 + t][lane * 2];
            pacc[t] = fmaf(wval.x, ival.x, fmaf(wval.y, ival.y, pacc[t]));
        }

        // All waves' LDS reads of buf[cur] retired (register deps force the
        // dscnt waits before this barrier), so buf[cur] is safe to refill.
        __syncthreads();
        if (c + 2 < NCH) stage(c + 2, cur);
    }

    // ---- reduce pmt partials across the wave (wave32) and write pmt/pnt ----
    #pragma unroll
    for (int t = 0; t < 16; ++t) {
        float s = pacc[t];
        s += __shfl_xor(s, 16, 32);
        s += __shfl_xor(s, 8, 32);
        s += __shfl_xor(s, 4, 32);
        s += __shfl_xor(s, 2, 32);
        s += __shfl_xor(s, 1, 32);
        if (lane == 0) {
            const int tok = tokBaseW + t;
            pmt[tok] = __expf(s + bias[tgt[t]]);
            pnt[tok] = uni[tgt[t]];
        }
    }

    // ---- epilogue: exp(acc + bias) -> pmn, broadcast unigram -> pnn ----
    // C/D layout: VGPR r holds (M = r + 8*half, N = lane%16) per 16x16 tile.
    #pragma unroll
    for (int r = 0; r < 8; ++r) {
        const int M   = r + 8 * half;
        const int tok = tokBaseW + M;
        const size_t base = (size_t)tok * KNZ;
        pmn[base + c0] = __expf(acc0[r] + bn0);
        pnn[base + c0] = un0;
        if (c1 < KNZ) {
            pmn[base + c1] = __expf(acc1[r] + bn1);
            pnn[base + c1] = un1;
        }
    }
}

extern "C" void kernel_launch(void* const* d_in, const int* in_sizes, int n_in,
                              void* d_out, int out_size, void* d_ws, size_t ws_size,
                              hipStream_t stream) {
    (void)in_sizes; (void)n_in; (void)out_size; (void)d_ws; (void)ws_size;
    const float* input  = (const float*)d_in[0];
    const float* weight = (const float*)d_in[1];
    const float* bias   = (const float*)d_in[2];
    const float* uni    = (const float*)d_in[3];
    const int*   target = (const int*)d_in[4];
    const int*   noise  = (const int*)d_in[5];

    float* out = (float*)d_out;
    float* pmt = out;
    float* pnt = out + N_TOK;
    float* pmn = out + 2 * (size_t)N_TOK;
    float* pnn = out + 2 * (size_t)N_TOK + (size_t)N_TOK * KNZ;

    dim3 grid(N_TOK / TM);   // 256 blocks
    dim3 block(BLK);         // 128 threads = 4 waves
    hipLaunchKernelGGL(nce_fused_kernel, grid, block, 0, stream,
                       input, weight, bias, uni, target, noise,
                       pmt, pnt, pmn, pnn);
}